// CenterLoss_8151847928313
// MI455X (gfx1250) — compile-verified
//
#include <hip/hip_runtime.h>
#include <hip/hip_bf16.h>
#include <math.h>

typedef __attribute__((ext_vector_type(2))) float v2f;
typedef __attribute__((ext_vector_type(8))) float v8f;

#define DIMS 128
#define TILE 16

// ---------------- workspace init: zero 2 doubles (S0,S1) + 1 u64 (count0) ---
__global__ void cl_init_ws(unsigned long long* ws) {
  if (threadIdx.x < 3) ws[threadIdx.x] = 0ull;
}

// ---------------- main pass: per-class distance sums via WMMA Gram diagonal -
__global__ __launch_bounds__(256) void cl_main(const float* __restrict__ f,
                                               const float* __restrict__ center,
                                               const int* __restrict__ t,
                                               double* __restrict__ s_ws,
                                               unsigned long long* __restrict__ c_ws,
                                               int n_rows, int n_tiles) {
  __shared__ float c_lds[2 * DIMS];
  for (int i = threadIdx.x; i < 2 * DIMS; i += blockDim.x) c_lds[i] = center[i];
  __syncthreads();

  const int lane = threadIdx.x & 31;
  const int wid  = blockIdx.x * (blockDim.x >> 5) + (threadIdx.x >> 5);
  const int nw   = gridDim.x * (blockDim.x >> 5);

  // WMMA 16x16x4 f32 A-operand mapping: this lane owns row (lane&15) of the
  // tile; lanes 16..31 carry the K+2/K+3 pair (byte offset +8).
  const int r    = lane & 15;
  const int koff = (lane >> 4) << 1;  // 0 or 2

  double s0 = 0.0, s1 = 0.0;
  unsigned int cnt0 = 0;

  for (int tile = wid; tile < n_tiles; tile += nw) {
    const int row = tile * TILE + r;
    const int cls = __builtin_nontemporal_load(t + row);
    const float* frow = f + (size_t)row * DIMS + koff;
    const float* crow = c_lds + cls * DIMS + koff;

    v8f acc = {};
#pragma unroll 4
    for (int k = 0; k < DIMS; k += 4) {
      // f is a read-once 512MB stream: non-temporal (TH=NT) load.
      v2f fv = __builtin_nontemporal_load((const v2f*)(frow + k));
      v2f cv = *(const v2f*)(crow + k);   // ds b64 from 1KB center copy
      v2f g;
      g.x = fv.x - cv.x;
      g.y = fv.y - cv.y;
      // B = A^T has the identical register image for 16x16x4 f32, so passing
      // g as both A and B accumulates the Gram matrix; diag = sum(g^2) per row.
      acc = __builtin_amdgcn_wmma_f32_16x16x4_f32(
          false, g, false, g, (short)0, acc, false, false);
    }

    // Diagonal extraction: element (m,m) sits at lane m, VGPR m (m=0..7) and
    // lane 24+j, VGPR j=m-8 (m=8..15).
    const bool lo = (lane < 8);
    const bool hi = (lane >= 24);
    const int  j  = lo ? lane : (lane - 24);
    float v = acc[0];
    v = (j == 1) ? acc[1] : v;
    v = (j == 2) ? acc[2] : v;
    v = (j == 3) ? acc[3] : v;
    v = (j == 4) ? acc[4] : v;
    v = (j == 5) ? acc[5] : v;
    v = (j == 6) ? acc[6] : v;
    v = (j == 7) ? acc[7] : v;
    if (lo || hi) {
      float dd = sqrtf(v);
      if (cls == 0) { s0 += (double)dd; cnt0++; }
      else          { s1 += (double)dd; }
    }
  }

  // Scalar tail for rows beyond the last full 16-row tile (none when N%16==0).
  for (int row = n_tiles * TILE + wid; row < n_rows; row += nw) {
    const int cls = t[row];
    const float* frow = f + (size_t)row * DIMS + lane * 4;
    const float* crow = c_lds + cls * DIMS + lane * 4;
    float p = 0.0f;
#pragma unroll
    for (int u = 0; u < 4; ++u) {
      float g = frow[u] - crow[u];
      p = fmaf(g, g, p);
    }
#pragma unroll
    for (int off = 16; off >= 1; off >>= 1) p += __shfl_down(p, off);
    if (lane == 0) {
      float dd = sqrtf(p);
      if (cls == 0) { s0 += (double)dd; cnt0++; }
      else          { s1 += (double)dd; }
    }
  }

  // Wave reduction, then one atomic per wave.
#pragma unroll
  for (int off = 16; off >= 1; off >>= 1) {
    s0   += __shfl_down(s0, off);
    s1   += __shfl_down(s1, off);
    cnt0 += __shfl_down(cnt0, off);
  }
  if (lane == 0) {
    atomicAdd(&s_ws[0], s0);
    atomicAdd(&s_ws[1], s1);
    atomicAdd(&c_ws[0], (unsigned long long)cnt0);
  }
}

// ---------------- finalize: out = S0/h0 + S1/h1 -----------------------------
__global__ void cl_finalize(const double* s_ws, const unsigned long long* c_ws,
                            float* out, int n_rows) {
  if (threadIdx.x == 0 && blockIdx.x == 0) {
    double s0 = s_ws[0], s1 = s_ws[1];
    double h0 = (double)c_ws[0];
    double h1 = (double)n_rows - h0;
    double res = 0.0;
    if (h0 > 0.0) res += s0 / h0;
    if (h1 > 0.0) res += s1 / h1;
    out[0] = (float)res;
  }
}

extern "C" void kernel_launch(void* const* d_in, const int* in_sizes, int n_in,
                              void* d_out, int out_size, void* d_ws, size_t ws_size,
                              hipStream_t stream) {
  const float* f      = (const float*)d_in[0];
  const float* center = (const float*)d_in[1];
  const int*   t      = (const int*)d_in[2];
  float* out = (float*)d_out;

  double*             s_ws = (double*)d_ws;
  unsigned long long* c_ws = (unsigned long long*)d_ws + 2;

  const int n_rows  = in_sizes[0] / DIMS;   // 1,000,000
  const int n_tiles = n_rows / TILE;        // 62,500

  cl_init_ws<<<1, 64, 0, stream>>>((unsigned long long*)d_ws);
  // 4096 blocks x 8 waves = 32768 waves: enough waves per SIMD32 to hide HBM
  // latency on a pure streaming pass (~1.9 tiles of 8KB per wave).
  cl_main<<<4096, 256, 0, stream>>>(f, center, t, s_ws, c_ws, n_rows, n_tiles);
  cl_finalize<<<1, 64, 0, stream>>>(s_ws, c_ws, out, n_rows);
}